// PSSM_46583215293003
// MI455X (gfx1250) — compile-verified
//
#include <hip/hip_runtime.h>
#include <cmath>

typedef float v2f __attribute__((ext_vector_type(2)));
typedef float v8f __attribute__((ext_vector_type(8)));

#define D_MODEL  1024
#define D_STATE  16
#define D_CONV   4
#define D_INNER  2048
#define DT_RANK  64
#define BATCH    4
#define SEQLEN   1024
#define BL       (BATCH * SEQLEN)   // 4096

// ---------------------------------------------------------------------------
// Generic fp32 WMMA GEMM:  C[M,N] = A[M,K] * W[N,K]^T   (all row-major fp32)
// One wave computes a 16 x (16*NT) strip using V_WMMA_F32_16X16X4_F32.
// Software pipeline: two *named* fragment sets (f0/f1, no runtime buffer
// index -> stays in VGPRs), K advanced two 16-k chunks per iteration so
// loads for one chunk overlap the 16 WMMAs of the other.
// K is a template parameter: all trip counts compile-time, K/16 even for
// every instantiation (64, 128, 4, 128).
// EPI==1 fuses: val = min(softplus(val + bias[n]) + 1e-4, 20)   (dt_proj)
// Per-lane fragment mapping (wave32, ISA 7.12.2):
//   A: lane holds A[m0 + lane%16][k0 + 2*(lane/16) + v], v=0..1
//   B: lane holds B[k0 + 2*(lane/16) + v][n0 + lane%16], v=0..1  (B[k][n]=W[n][k])
//   C: lane holds C[m0 + v + 8*(lane/16)][n0 + lane%16], v=0..7
// ---------------------------------------------------------------------------
template <int NT, int EPI, int K>
__global__ __launch_bounds__(128) void gemm_wt_kernel(
    const float* __restrict__ A, int lda,
    const float* __restrict__ W, int ldw,
    const float* __restrict__ bias,
    float* __restrict__ C, int ldc)
{
    const int lane = threadIdx.x & 31;
    const int wave = threadIdx.x >> 5;
    const int r    = lane & 15;
    const int hk   = lane >> 4;
    const int m0   = (blockIdx.y * 4 + wave) * 16;
    const int n0   = blockIdx.x * (16 * NT);

    v8f acc[NT] = {};

    const float* arow = A + (size_t)(m0 + r) * lda;
    const float* wrow[NT];
#pragma unroll
    for (int j = 0; j < NT; ++j)
        wrow[j] = W + (size_t)(n0 + j * 16 + r) * ldw;

    constexpr int CH  = 4;           // wmma steps per chunk (16 k-values)
    constexpr int NCH = K / (CH * 4);
    static_assert(NCH >= 2 && (NCH % 2) == 0, "K must give an even chunk count");

    struct Frag {
        v2f a[CH];
        v2f b[NT][CH];
    };
    Frag f0, f1;

    auto load_chunk = [&](int kc, Frag& f) {
        const int kbase = kc * (CH * 4) + 2 * hk;
#pragma unroll
        for (int s = 0; s < CH; ++s) {
            const int ka = kbase + 4 * s;
            v2f a;
            a.x = arow[ka];
            a.y = arow[ka + 1];
            f.a[s] = a;
#pragma unroll
            for (int j = 0; j < NT; ++j) {
                v2f b;
                b.x = wrow[j][ka];
                b.y = wrow[j][ka + 1];
                f.b[j][s] = b;
            }
        }
    };

    auto compute_chunk = [&](Frag& f) {
#pragma unroll
        for (int s = 0; s < CH; ++s) {
#pragma unroll
            for (int j = 0; j < NT; ++j) {
                acc[j] = __builtin_amdgcn_wmma_f32_16x16x4_f32(
                    false, f.a[s], false, f.b[j][s],
                    (short)0, acc[j], false, false);
            }
        }
    };

    load_chunk(0, f0);
    for (int kc = 0; kc < NCH; kc += 2) {
        load_chunk(kc + 1, f1);          // NCH even -> kc+1 always valid
        compute_chunk(f0);
        if (kc + 2 < NCH) load_chunk(kc + 2, f0);
        compute_chunk(f1);
    }

#pragma unroll
    for (int j = 0; j < NT; ++j) {
        const int n = n0 + j * 16 + r;
#pragma unroll
        for (int v = 0; v < 8; ++v) {
            const int m = m0 + v + 8 * hk;
            float val = acc[j][v];
            if (EPI == 1) {
                val += bias[n];
                float sp = (val > 20.f) ? val : log1pf(expf(val));
                val = fminf(sp + 1e-4f, 20.f);
            }
            C[(size_t)m * ldc + n] = val;
        }
    }
}

// ---------------------------------------------------------------------------
// Causal depthwise conv1d (k=4, left pad 3) + bias + SiLU.
// xc lives in the first half of xz rows: xz[bl*4096 + d], d < 2048.
// ---------------------------------------------------------------------------
__global__ __launch_bounds__(256) void conv_silu_kernel(
    const float* __restrict__ xz,
    const float* __restrict__ conv_w,   // [D_INNER, 1, 4]
    const float* __restrict__ conv_b,   // [D_INNER]
    float* __restrict__ u)              // [BL, D_INNER]
{
    int idx = blockIdx.x * blockDim.x + threadIdx.x;  // over BL*D_INNER
    int d   = idx & (D_INNER - 1);
    int bl  = idx >> 11;                              // /2048
    int l   = bl & (SEQLEN - 1);

    float s = conv_b[d];
#pragma unroll
    for (int t = 0; t < D_CONV; ++t) {
        int lp = l - (D_CONV - 1) + t;
        if (lp >= 0)
            s += xz[(size_t)(bl - (D_CONV - 1) + t) * (2 * D_INNER) + d] *
                 conv_w[d * D_CONV + t];
    }
    float sig = 1.f / (1.f + expf(-s));
    u[(size_t)idx] = s * sig;
}

// ---------------------------------------------------------------------------
// Selective scan.  Block = 256 threads = 16 channels (d) x 16 states (n).
// h is register-resident; y reduced over n with width-16 shuffles.
// Time axis processed in chunks of T=8: all loads for a chunk issue up front
// (amortizing L2 latency 8x), and the next chunk is prefetched
// (global_prefetch_b8) while the exp/fma recurrence of this chunk runs.
// Fuses  y = (sum_n h*C) + u*D, then y *= silu(z).
// ---------------------------------------------------------------------------
__global__ __launch_bounds__(256) void scan_kernel(
    const float* __restrict__ delta,   // [BL, D_INNER]
    const float* __restrict__ xdbl,    // [BL, 96]  (dt | B | C)
    const float* __restrict__ u,       // [BL, D_INNER]
    const float* __restrict__ xz,      // [BL, 2*D_INNER] (z = cols 2048..4095)
    const float* __restrict__ A_log,   // [D_INNER, D_STATE]
    const float* __restrict__ Dp,      // [D_INNER]
    float* __restrict__ ypost)         // [BL, D_INNER]
{
    const int n  = threadIdx.x & 15;
    const int dl = threadIdx.x >> 4;           // 0..15
    const int d  = blockIdx.x * 16 + dl;
    const int b  = blockIdx.y;

    const float Ad = -expf(fminf(A_log[d * D_STATE + n], 10.f));
    const float Dd = Dp[d];

    float h = 0.f;
    const int base_bl = b * SEQLEN;

    constexpr int T = 8;
    for (int l0 = 0; l0 < SEQLEN; l0 += T) {
        float dv[T], uv[T], Bv[T], Cv[T];
#pragma unroll
        for (int t = 0; t < T; ++t) {
            const int bl = base_bl + l0 + t;
            dv[t] = delta[(size_t)bl * D_INNER + d];
            uv[t] = u[(size_t)bl * D_INNER + d];
            Bv[t] = xdbl[(size_t)bl * 96 + DT_RANK + n];
            Cv[t] = xdbl[(size_t)bl * 96 + DT_RANK + D_STATE + n];
        }
        if (l0 + T < SEQLEN) {
            const int bln = base_bl + l0 + T;
            __builtin_prefetch(&delta[(size_t)bln * D_INNER + d], 0, 1);
            __builtin_prefetch(&u[(size_t)bln * D_INNER + d], 0, 1);
            __builtin_prefetch(&xdbl[(size_t)bln * 96 + DT_RANK + n], 0, 1);
            __builtin_prefetch(&xz[(size_t)bln * (2 * D_INNER) + D_INNER + d], 0, 1);
        }
#pragma unroll
        for (int t = 0; t < T; ++t) {
            const float dA = expf(dv[t] * Ad);
            h = dA * h + dv[t] * Bv[t] * uv[t];

            float p = h * Cv[t];
            p += __shfl_xor(p, 8, 16);
            p += __shfl_xor(p, 4, 16);
            p += __shfl_xor(p, 2, 16);
            p += __shfl_xor(p, 1, 16);

            if (n == 0) {
                const int bl = base_bl + l0 + t;
                const float zv  = xz[(size_t)bl * (2 * D_INNER) + D_INNER + d];
                const float sil = zv / (1.f + expf(-zv));
                ypost[(size_t)bl * D_INNER + d] = (p + uv[t] * Dd) * sil;
            }
        }
    }
}

// ---------------------------------------------------------------------------
extern "C" void kernel_launch(void* const* d_in, const int* in_sizes, int n_in,
                              void* d_out, int out_size, void* d_ws, size_t ws_size,
                              hipStream_t stream)
{
    const float* x         = (const float*)d_in[0];  // [4,1024,1024]
    const float* in_proj_w = (const float*)d_in[1];  // [4096,1024]
    const float* conv_w    = (const float*)d_in[2];  // [2048,1,4]
    const float* conv_b    = (const float*)d_in[3];  // [2048]
    const float* x_proj_w  = (const float*)d_in[4];  // [96,2048]
    const float* dt_proj_w = (const float*)d_in[5];  // [2048,64]
    const float* dt_proj_b = (const float*)d_in[6];  // [2048]
    const float* A_log     = (const float*)d_in[7];  // [2048,16]
    const float* Dp        = (const float*)d_in[8];  // [2048]
    const float* out_proj_w= (const float*)d_in[9];  // [1024,2048]
    float* out = (float*)d_out;                      // [4,1024,1024]

    // workspace layout (fp32)
    float* xz    = (float*)d_ws;                   // [BL, 4096]
    float* u     = xz    + (size_t)BL * 4096;      // [BL, 2048]
    float* xdbl  = u     + (size_t)BL * 2048;      // [BL, 96]
    float* delta = xdbl  + (size_t)BL * 96;        // [BL, 2048]
    float* ypost = delta + (size_t)BL * 2048;      // [BL, 2048]

    // 1) in_proj: xz[BL,4096] = x[BL,1024] @ in_proj_w^T      (M=4096,N=4096,K=1024)
    gemm_wt_kernel<4, 0, D_MODEL><<<dim3(4096 / 64, BL / 64), 128, 0, stream>>>(
        x, D_MODEL, in_proj_w, D_MODEL, nullptr, xz, 2 * D_INNER);

    // 2) depthwise causal conv + SiLU -> u[BL,2048]
    conv_silu_kernel<<<(BL * D_INNER) / 256, 256, 0, stream>>>(xz, conv_w, conv_b, u);

    // 3) x_proj: xdbl[BL,96] = u @ x_proj_w^T                 (N=96 -> NT=6, one strip)
    gemm_wt_kernel<6, 0, D_INNER><<<dim3(1, BL / 64), 128, 0, stream>>>(
        u, D_INNER, x_proj_w, D_INNER, nullptr, xdbl, 96);

    // 4) dt_proj + softplus/clip: delta[BL,2048]              (K=64, A = xdbl[:, :64])
    gemm_wt_kernel<4, 1, DT_RANK><<<dim3(D_INNER / 64, BL / 64), 128, 0, stream>>>(
        xdbl, 96, dt_proj_w, DT_RANK, dt_proj_b, delta, D_INNER);

    // 5) selective scan + gating -> ypost[BL,2048]
    scan_kernel<<<dim3(D_INNER / 16, BATCH), 256, 0, stream>>>(
        delta, xdbl, u, xz, A_log, Dp, ypost);

    // 6) out_proj: out[BL,1024] = ypost @ out_proj_w^T        (M=4096,N=1024,K=2048)
    gemm_wt_kernel<4, 0, D_INNER><<<dim3(D_MODEL / 64, BL / 64), 128, 0, stream>>>(
        ypost, D_INNER, out_proj_w, D_INNER, nullptr, out, D_MODEL);
}